// GCN_12661563589059
// MI455X (gfx1250) — compile-verified
//
#include <hip/hip_runtime.h>
#include <hip/hip_bf16.h>

typedef __attribute__((ext_vector_type(2))) float v2f;
typedef __attribute__((ext_vector_type(8))) float v8f;

#define IN_FEATS   128
#define OUT_FEATS  128
#define TILE_M     16
#define LDS_STRIDE 132   // 128 + 4: rows land on distinct bank groups (132 mod 64 = 4)

__global__ __launch_bounds__(256)
void gcn_fused_kernel(const float* __restrict__ x,
                      const int*   __restrict__ rowptr,
                      const int*   __restrict__ cols,
                      const float* __restrict__ vals,
                      const float* __restrict__ W,
                      const float* __restrict__ bias,
                      float*       __restrict__ out,
                      int n_nodes)
{
    __shared__ float agg[TILE_M * LDS_STRIDE];   // 16 x 128 f32 tile, padded

    const int tid  = threadIdx.x;
    const int lane = tid & 31;
    const int wave = tid >> 5;          // 0..7
    const int row0 = blockIdx.x * TILE_M;

    // ---------------- Phase 1: CSR gather + scale + aggregate -> LDS ----------
    // wave handles nodes (row0 + 2*wave + {0,1}); lane owns feats [4*lane, 4*lane+4)
    for (int s = 0; s < 2; ++s) {
        const int nl   = 2 * wave + s;          // 0..15 within tile
        const int node = row0 + nl;
        float4 acc = make_float4(0.f, 0.f, 0.f, 0.f);
        if (node < n_nodes) {
            const int start = rowptr[node];
            const int end   = rowptr[node + 1];
            for (int e0 = start; e0 < end; e0 += 32) {
                const int e = e0 + lane;
                int   c = 0;
                float v = 0.f;
                if (e < end) { c = cols[e]; v = vals[e]; }   // one coalesced edge load per lane
                const int cnt = min(32, end - e0);
                for (int j = 0; j < cnt; ++j) {
                    const int   cj = __shfl(c, j, 32);       // wave32 broadcast
                    const float vj = __shfl(v, j, 32);
                    const float4 xr = *(const float4*)(x + (size_t)cj * IN_FEATS + lane * 4);
                    acc.x += vj * xr.x;  acc.y += vj * xr.y;
                    acc.z += vj * xr.z;  acc.w += vj * xr.w;
                }
            }
        }
        float* dst = &agg[nl * LDS_STRIDE + lane * 4];
        dst[0] = acc.x; dst[1] = acc.y; dst[2] = acc.z; dst[3] = acc.w;
    }
    __syncthreads();

    // ---------------- Phase 2: [16x128] @ [128x16] via V_WMMA_F32_16X16X4_F32 --
    // Wave w computes output columns [16w, 16w+16).
    const int m     = lane & 15;             // A row / C column index
    const int kbase = (lane >> 4) << 1;      // 0 for lanes 0-15, 2 for lanes 16-31
    const int ncol  = (wave << 4) + m;       // output column for B/C/D fragments

    // Preload B panel (K=128 x N=16) into registers: 32 k-steps of the 4-wide slice.
    v2f b[32];
#pragma unroll
    for (int kk = 0; kk < 32; ++kk) {
        const int k = 4 * kk + kbase;
        v2f t;
        t.x = W[(size_t)k       * OUT_FEATS + ncol];
        t.y = W[(size_t)(k + 1) * OUT_FEATS + ncol];
        b[kk] = t;
    }

    // Accumulator seeded with bias (free epilogue).
    v8f c;
    const float bv = bias[ncol];
#pragma unroll
    for (int r = 0; r < 8; ++r) c[r] = bv;

#pragma unroll
    for (int kk = 0; kk < 32; ++kk) {
        const int k = 4 * kk + kbase;
        v2f a;
        a.x = agg[m * LDS_STRIDE + k];
        a.y = agg[m * LDS_STRIDE + k + 1];
        // D = A x B + C   (8 args: neg_a, A, neg_b, B, c_mod, C, reuse_a, reuse_b)
        c = __builtin_amdgcn_wmma_f32_16x16x4_f32(false, a, false, b[kk],
                                                  (short)0, c, false, false);
    }

    // Store: reg r -> tile row (r + 8*half), column ncol.
    const int half = lane >> 4;
#pragma unroll
    for (int r = 0; r < 8; ++r) {
        const int row = row0 + r + 8 * half;
        if (row < n_nodes) out[(size_t)row * OUT_FEATS + ncol] = c[r];
    }
}

extern "C" void kernel_launch(void* const* d_in, const int* in_sizes, int n_in,
                              void* d_out, int out_size, void* d_ws, size_t ws_size,
                              hipStream_t stream) {
    const float* x      = (const float*)d_in[0];   // input_dense [N,128]
    const int*   rowptr = (const int*)  d_in[1];   // offset_graph [N+1]
    const int*   cols   = (const int*)  d_in[2];   // cols_graph [E]
    const float* vals   = (const float*)d_in[3];   // vals_graph [E]
    const float* W      = (const float*)d_in[4];   // weights [128,128]
    const float* bias   = (const float*)d_in[5];   // bias [128]
    float*       out    = (float*)d_out;           // [N,128]

    const int n_nodes = in_sizes[0] / IN_FEATS;
    const int grid    = (n_nodes + TILE_M - 1) / TILE_M;

    gcn_fused_kernel<<<grid, 256, 0, stream>>>(x, rowptr, cols, vals, W, bias,
                                               out, n_nodes);
}